// OneStep_27822798143729
// MI455X (gfx1250) — compile-verified
//
#include <hip/hip_runtime.h>

// CDNA5 / gfx1250: wave32, f32 WMMA 16x16x4 path (matches reference f32 numerics).
typedef __attribute__((ext_vector_type(2))) float v2f;
typedef __attribute__((ext_vector_type(8))) float v8f;

#define D   4096
#define CTX 512
// ALPHA = LR * 2 / (CTX*D) = 0.1 * 2 / 2097152
#define ALPHA (0.2f / (float)((size_t)CTX * (size_t)D))

#define TM 128            // M tile per block
#define TN 64             // N tile per block
#define TK 32             // K stage depth
#define APITCH (TK + 4)   // 36 dwords: 36*l mod 64 hits all 4k residues -> conflict-free A b64 reads
#define BPITCH (TN + 16)  // 80 float2 = 160 dwords = +32 banks/row -> conflict-free B b64 reads

// ---------------------------------------------------------------------------
// Phase 1: E = keys @ W - vals      (M=CTX, N=D, K=D)
// ---------------------------------------------------------------------------
__global__ __launch_bounds__(256) void gemm_e_kernel(
    const float* __restrict__ keys, const float* __restrict__ W,
    const float* __restrict__ vals, float* __restrict__ E)
{
    __shared__ float  As[2][TM][APITCH];       // [m][k]
    __shared__ float2 Bs[2][TK / 2][BPITCH];   // [k/2][n] = {B[k][n], B[k+1][n]}

    const int tid  = threadIdx.x;
    const int wave = tid >> 5, lane = tid & 31;
    const int l = lane & 15, hi = lane >> 4, ksel = hi << 1;
    const int wm = (wave & 3) << 5;            // 4 waves along M: 32-row strips
    const int wn = (wave >> 2) << 4 << 1;      // 2 waves along N: 32-col strips
    const int mBlk = blockIdx.y * TM, nBlk = blockIdx.x * TN;

    v8f acc[2][2] = {};
    float4 rA[4], rB[2];

    auto gload = [&](int kb) {                 // global -> regs (next stage)
#pragma unroll
        for (int i = 0; i < 4; ++i) {          // A: 128x32, 8 float4/row
            int f = tid + (i << 8);
            rA[i] = *(const float4*)&keys[(size_t)(mBlk + (f >> 3)) * D + kb + ((f & 7) << 2)];
        }
#pragma unroll
        for (int i = 0; i < 2; ++i) {          // B: 32x64, 16 float4/row
            int f = tid + (i << 8);
            rB[i] = *(const float4*)&W[(size_t)(kb + (f >> 4)) * D + nBlk + ((f & 15) << 2)];
        }
    };
    auto lstore = [&](int buf) {               // regs -> LDS
#pragma unroll
        for (int i = 0; i < 4; ++i) {
            int f = tid + (i << 8);
            int row = f >> 3, c = (f & 7) << 2;
            As[buf][row][c + 0] = rA[i].x; As[buf][row][c + 1] = rA[i].y;
            As[buf][row][c + 2] = rA[i].z; As[buf][row][c + 3] = rA[i].w;
        }
#pragma unroll
        for (int i = 0; i < 2; ++i) {          // k-pair interleave
            int f = tid + (i << 8);
            int row = f >> 4, c = (f & 15) << 2, o = row & 1;
            float* p = (float*)&Bs[buf][row >> 1][c];
            p[0 * 2 + o] = rB[i].x; p[1 * 2 + o] = rB[i].y;
            p[2 * 2 + o] = rB[i].z; p[3 * 2 + o] = rB[i].w;
        }
    };

    gload(0);
    lstore(0);
    __syncthreads();

    for (int kb = 0; kb < D; kb += TK) {
        const int buf = (kb >> 5) & 1;
        const bool more = (kb + TK) < D;
        if (more) gload(kb + TK);              // overlap global loads with WMMA

#pragma unroll
        for (int kk = 0; kk < TK; kk += 4) {
            v2f a[2], b[2];
#pragma unroll
            for (int s = 0; s < 2; ++s)
                a[s] = *(const v2f*)&As[buf][wm + (s << 4) + l][kk + ksel];
#pragma unroll
            for (int t = 0; t < 2; ++t)
                b[t] = *(const v2f*)&Bs[buf][(kk >> 1) + hi][wn + (t << 4) + l];
#pragma unroll
            for (int s = 0; s < 2; ++s)
#pragma unroll
                for (int t = 0; t < 2; ++t)
                    acc[s][t] = __builtin_amdgcn_wmma_f32_16x16x4_f32(
                        false, a[s], false, b[t], (short)0, acc[s][t], false, false);
        }

        if (more) lstore(buf ^ 1);
        __syncthreads();
    }

    // fused epilogue: E = acc - vals
#pragma unroll
    for (int s = 0; s < 2; ++s)
#pragma unroll
        for (int t = 0; t < 2; ++t)
#pragma unroll
            for (int r = 0; r < 8; ++r) {
                int row = mBlk + wm + (s << 4) + r + (hi << 3);
                int col = nBlk + wn + (t << 4) + l;
                size_t idx = (size_t)row * D + col;
                E[idx] = acc[s][t][r] - vals[idx];
            }
}

// ---------------------------------------------------------------------------
// Phase 2: Wout = W - ALPHA * keys^T @ E      (M=D, N=D, K=CTX)
//   A[m][k] = keys[k*D + m]  -> transpose during regs->LDS stage
// ---------------------------------------------------------------------------
__global__ __launch_bounds__(256) void gemm_update_kernel(
    const float* __restrict__ keys, const float* __restrict__ E,
    const float* __restrict__ W, float* __restrict__ Wout)
{
    __shared__ float  As[2][TM][APITCH];
    __shared__ float2 Bs[2][TK / 2][BPITCH];

    const int tid  = threadIdx.x;
    const int wave = tid >> 5, lane = tid & 31;
    const int l = lane & 15, hi = lane >> 4, ksel = hi << 1;
    const int wm = (wave & 3) << 5;
    const int wn = (wave >> 2) << 4 << 1;
    const int mBlk = blockIdx.y * TM, nBlk = blockIdx.x * TN;

    v8f acc[2][2] = {};
    float4 rA[4], rB[2];

    auto gload = [&](int kb) {
#pragma unroll
        for (int i = 0; i < 4; ++i) {          // keys rows k (coalesced): 32x128
            int f = tid + (i << 8);
            rA[i] = *(const float4*)&keys[(size_t)(kb + (f >> 5)) * D + mBlk + ((f & 31) << 2)];
        }
#pragma unroll
        for (int i = 0; i < 2; ++i) {          // B from E: 32x64
            int f = tid + (i << 8);
            rB[i] = *(const float4*)&E[(size_t)(kb + (f >> 4)) * D + nBlk + ((f & 15) << 2)];
        }
    };
    auto lstore = [&](int buf) {
#pragma unroll
        for (int i = 0; i < 4; ++i) {          // transpose: As[m][k] = keys[k][m]
            int f = tid + (i << 8);
            int srow = f >> 5, c = (f & 31) << 2;
            As[buf][c + 0][srow] = rA[i].x; As[buf][c + 1][srow] = rA[i].y;
            As[buf][c + 2][srow] = rA[i].z; As[buf][c + 3][srow] = rA[i].w;
        }
#pragma unroll
        for (int i = 0; i < 2; ++i) {
            int f = tid + (i << 8);
            int row = f >> 4, c = (f & 15) << 2, o = row & 1;
            float* p = (float*)&Bs[buf][row >> 1][c];
            p[0 * 2 + o] = rB[i].x; p[1 * 2 + o] = rB[i].y;
            p[2 * 2 + o] = rB[i].z; p[3 * 2 + o] = rB[i].w;
        }
    };

    gload(0);
    lstore(0);
    __syncthreads();

    for (int kb = 0; kb < CTX; kb += TK) {
        const int buf = (kb >> 5) & 1;
        const bool more = (kb + TK) < CTX;
        if (more) gload(kb + TK);

#pragma unroll
        for (int kk = 0; kk < TK; kk += 4) {
            v2f a[2], b[2];
#pragma unroll
            for (int s = 0; s < 2; ++s)
                a[s] = *(const v2f*)&As[buf][wm + (s << 4) + l][kk + ksel];
#pragma unroll
            for (int t = 0; t < 2; ++t)
                b[t] = *(const v2f*)&Bs[buf][(kk >> 1) + hi][wn + (t << 4) + l];
#pragma unroll
            for (int s = 0; s < 2; ++s)
#pragma unroll
                for (int t = 0; t < 2; ++t)
                    acc[s][t] = __builtin_amdgcn_wmma_f32_16x16x4_f32(
                        false, a[s], false, b[t], (short)0, acc[s][t], false, false);
        }

        if (more) lstore(buf ^ 1);
        __syncthreads();
    }

    // fused epilogue: Wout = W - ALPHA * acc
#pragma unroll
    for (int s = 0; s < 2; ++s)
#pragma unroll
        for (int t = 0; t < 2; ++t)
#pragma unroll
            for (int r = 0; r < 8; ++r) {
                int row = mBlk + wm + (s << 4) + r + (hi << 3);
                int col = nBlk + wn + (t << 4) + l;
                size_t idx = (size_t)row * D + col;
                Wout[idx] = W[idx] - ALPHA * acc[s][t][r];
            }
}

// ---------------------------------------------------------------------------
extern "C" void kernel_launch(void* const* d_in, const int* in_sizes, int n_in,
                              void* d_out, int out_size, void* d_ws, size_t ws_size,
                              hipStream_t stream) {
    // setup_inputs() order: W (D*D), keys (CTX*D), vals (CTX*D), all float32
    const float* W    = (const float*)d_in[0];
    const float* keys = (const float*)d_in[1];
    const float* vals = (const float*)d_in[2];
    float*       out  = (float*)d_out;          // new W, D*D float32
    float*       Ebuf = (float*)d_ws;           // CTX*D float32 scratch (8 MB)

    dim3 block(256);
    // Phase 1: E = keys @ W - vals
    dim3 grid1(D / TN, CTX / TM);
    gemm_e_kernel<<<grid1, block, 0, stream>>>(keys, W, vals, Ebuf);
    // Phase 2: out = W - ALPHA * keys^T @ E
    dim3 grid2(D / TN, D / TM);
    gemm_update_kernel<<<grid2, block, 0, stream>>>(keys, Ebuf, W, out);
}